// TripletModel_78091095376057
// MI455X (gfx1250) — compile-verified
//
#include <hip/hip_runtime.h>

typedef float v4f __attribute__((ext_vector_type(4)));
typedef float v2f __attribute__((ext_vector_type(2)));
typedef float v8f __attribute__((ext_vector_type(8)));

#define FEAT 384
#define ALPHA 0.2f
#define TPB 256
#define WAVES_PER_BLOCK (TPB / 32)
#define NBLOCKS 2048

// ---------------------------------------------------------------------------
// Kernel 1: bandwidth-bound streaming pass.
// One wave per row; lane l covers features l*4 + {0,128,256} (3x b128 loads
// per input array, 512B contiguous per wave per load). Non-temporal: the
// 604MB stream is larger than the 192MB L2, so don't pollute it.
// ---------------------------------------------------------------------------
__global__ __launch_bounds__(TPB) void triplet_partial_kernel(
    const float* __restrict__ A, const float* __restrict__ P,
    const float* __restrict__ N, const float* __restrict__ Wv,
    float* __restrict__ partial, int batch)
{
    __shared__ float s_w[FEAT];
    __shared__ float s_wave[WAVES_PER_BLOCK];

    const int tid = threadIdx.x;
    // exp(W) once per block into LDS
    for (int f = tid; f < FEAT; f += TPB) s_w[f] = expf(Wv[f]);
    __syncthreads();

    const int lane = tid & 31;
    const int wave = tid >> 5;
    const int col  = lane * 4;

    // Row-invariant weight slices, hoisted out of the row loop.
    const v4f w0 = *(const v4f*)&s_w[col];
    const v4f w1 = *(const v4f*)&s_w[col + 128];
    const v4f w2 = *(const v4f*)&s_w[col + 256];

    const int gwave = blockIdx.x * WAVES_PER_BLOCK + wave;
    const int nwave = gridDim.x * WAVES_PER_BLOCK;

    float wacc = 0.0f;
    for (int row = gwave; row < batch; row += nwave) {
        const size_t base = (size_t)row * FEAT + col;

        const v4f a0 = __builtin_nontemporal_load((const v4f*)(A + base));
        const v4f a1 = __builtin_nontemporal_load((const v4f*)(A + base + 128));
        const v4f a2 = __builtin_nontemporal_load((const v4f*)(A + base + 256));
        const v4f p0 = __builtin_nontemporal_load((const v4f*)(P + base));
        const v4f p1 = __builtin_nontemporal_load((const v4f*)(P + base + 128));
        const v4f p2 = __builtin_nontemporal_load((const v4f*)(P + base + 256));
        const v4f n0 = __builtin_nontemporal_load((const v4f*)(N + base));
        const v4f n1 = __builtin_nontemporal_load((const v4f*)(N + base + 128));
        const v4f n2 = __builtin_nontemporal_load((const v4f*)(N + base + 256));

        v4f dp0 = a0 - p0, dn0 = a0 - n0;
        v4f dp1 = a1 - p1, dn1 = a1 - n1;
        v4f dp2 = a2 - p2, dn2 = a2 - n2;

        v4f t = w0 * (dp0 * dp0 - dn0 * dn0)
              + w1 * (dp1 * dp1 - dn1 * dn1)
              + w2 * (dp2 * dp2 - dn2 * dn2);

        float acc = t.x + t.y + t.z + t.w;   // = D_pos - D_neg (lane partial)

        // 32-lane butterfly reduce (wave32)
        #pragma unroll
        for (int off = 16; off > 0; off >>= 1)
            acc += __shfl_xor(acc, off, 32);

        wacc += fmaxf(acc + ALPHA, 0.0f);    // relu, same value in all lanes
    }

    if (lane == 0) s_wave[wave] = wacc;
    __syncthreads();
    if (tid == 0) {
        float s = 0.0f;
        #pragma unroll
        for (int i = 0; i < WAVES_PER_BLOCK; ++i) s += s_wave[i];
        partial[blockIdx.x] = s;
    }
}

// ---------------------------------------------------------------------------
// Kernel 2: single-wave deterministic finalize using V_WMMA_F32_16X16X4_F32.
// Lane accumulators are placed in A (lanes 0-15 -> K=0, lanes 16-31 -> K=2,
// other K slots zero). B = ones reduces over K: D[m][n] = acc[m]+acc[m+16].
// Per the 16x16 f32 C/D layout, lane l<16 holds D[0..7][l] in d[0..7] and
// lane l>=16 holds D[8..15][l-16], so an 8-add + one shfl_xor(16) completes
// the exact f32 sum.
// ---------------------------------------------------------------------------
__global__ __launch_bounds__(32) void triplet_finalize_kernel(
    const float* __restrict__ partial, float* __restrict__ out,
    int count, float inv_batch)
{
    const int lane = threadIdx.x;

    float acc = 0.0f;
    for (int i = lane; i < count; i += 32)   // fixed order -> deterministic
        acc += partial[i];

    v2f a; a.x = acc;  a.y = 0.0f;
    v2f b; b.x = 1.0f; b.y = 1.0f;
    v8f c = {};
    // 8 args: (neg_a, A, neg_b, B, c_mod, C, reuse_a, reuse_b)
    v8f d = __builtin_amdgcn_wmma_f32_16x16x4_f32(
        false, a, false, b, (short)0, c, false, false);

    float s = d[0] + d[1] + d[2] + d[3] + d[4] + d[5] + d[6] + d[7];
    float tot = s + __shfl_xor(s, 16, 32);

    if (lane == 0) out[0] = tot * inv_batch;
}

// ---------------------------------------------------------------------------
extern "C" void kernel_launch(void* const* d_in, const int* in_sizes, int n_in,
                              void* d_out, int out_size, void* d_ws, size_t ws_size,
                              hipStream_t stream)
{
    const float* A = (const float*)d_in[0];   // anchor   [B, F] f32
    const float* P = (const float*)d_in[1];   // positive [B, F] f32
    const float* N = (const float*)d_in[2];   // negative [B, F] f32
    const float* W = (const float*)d_in[3];   // W        [F]    f32

    const int batch = in_sizes[0] / FEAT;

    int blocks = NBLOCKS;
    if ((size_t)blocks * sizeof(float) > ws_size)
        blocks = (int)(ws_size / sizeof(float));
    if (blocks < 1) blocks = 1;

    float* partial = (float*)d_ws;

    triplet_partial_kernel<<<blocks, TPB, 0, stream>>>(A, P, N, W, partial, batch);
    triplet_finalize_kernel<<<1, 32, 0, stream>>>(partial, (float*)d_out,
                                                  blocks, 1.0f / (float)batch);
}